// FLIF_25125558682111
// MI455X (gfx1250) — compile-verified
//
#include <hip/hip_runtime.h>
#include <math.h>

// FLIF scan, T=256, B=16, S=2048.  Fractional-memory term restructured as
// chunked f32 WMMA GEMMs (V_WMMA_F32_16X16X4_F32) over the LDS-resident
// delta history; intra-chunk triangular tail handled scalar per lane.

#define T_STEPS   256
#define BSIZE     16
#define LAYER     2048
#define BS        (BSIZE * LAYER)      // 32768 neurons
#define CHUNK     16
#define NCHUNK    (T_STEPS / CHUNK)    // 16

#define ALPHA_F        0.2f
#define DT_F           0.1f
#define THRESHOLD_F   (-50.0f)
#define V_INIT_F      (-70.0f)
#define VL_F          (-70.0f)
#define V_RESET_F     (-70.0f)
#define GL_F           0.025f
#define CM_F           0.5f
#define INV_GL_F       40.0f           // 1/GL
#define INV_CM_F       2.0f            // 1/CM
#define DT_OVER_TAU_F  0.005f          // DT / (CM/GL)
#define GAMMA_C_F      0.58766366f     // DT^ALPHA * Gamma(2-ALPHA)

typedef float v2f __attribute__((ext_vector_type(2)));
typedef float v8f __attribute__((ext_vector_type(8)));

__global__ __launch_bounds__(32)
void flif_scan_kernel(const float* __restrict__ I,
                      const float* __restrict__ V0,
                      float* __restrict__ spikes_out,
                      float* __restrict__ V_out) {
    // k[q] = (q+1)^(1-a) - q^(1-a); weight for (step n, history slot j) is k[n-j]
    __shared__ float ktab[T_STEPS];
    // delta history for the wave's two 16-neuron groups: [group][slot][neuron]
    __shared__ float d_lds[2][T_STEPS][16];
    // past-history memV tile for the current chunk: [group][row][neuron]
    __shared__ float memv_lds[2][CHUNK][16];

    const int lane = threadIdx.x;          // 0..31 (wave32)
    const int grp  = lane >> 4;            // which 16-neuron group this lane owns
    const int e16  = lane & 15;            // neuron-in-group / matrix column
    const int eg   = blockIdx.x * 32 + lane;  // global neuron index (b*S + s)

    // ---- init: weight table + zero history (pad slots must read as 0) ----
    for (int q = lane; q < T_STEPS; q += 32)
        ktab[q] = __powf((float)(q + 1), 1.0f - ALPHA_F) - __powf((float)q, 1.0f - ALPHA_F);
    {
        float* dz = &d_lds[0][0][0];
        for (int i = lane; i < 2 * T_STEPS * 16; i += 32) dz[i] = 0.0f;
    }
    __syncthreads();

    float V = V0[eg];

    for (int c = 0; c < NCHUNK; ++c) {
        const int n0 = c * CHUNK;

        // ===== GEMM phase: memV contribution of all pre-chunk history =====
        // A (16x4 f32): lane = row M, K pair {0,1} in lanes 0-15, {2,3} in 16-31
        // B (4x16 f32): lane = col N, same K-pair split across lane halves
        v8f accA = {0.f, 0.f, 0.f, 0.f, 0.f, 0.f, 0.f, 0.f};
        v8f accB = {0.f, 0.f, 0.f, 0.f, 0.f, 0.f, 0.f, 0.f};
        const int kkb = (lane < 16) ? 0 : 2;
        for (int j0 = 0; j0 < n0; j0 += 4) {
            // weights: A[r][kk] = k[(n0+r) - (j0+kk)]  (r = e16 for this lane)
            v2f a;
            a.x = ktab[n0 + e16 - (j0 + kkb)];
            a.y = ktab[n0 + e16 - (j0 + kkb + 1)];
            v2f bA, bB;
            bA.x = d_lds[0][j0 + kkb][e16];
            bA.y = d_lds[0][j0 + kkb + 1][e16];
            bB.x = d_lds[1][j0 + kkb][e16];
            bB.y = d_lds[1][j0 + kkb + 1][e16];
            accA = __builtin_amdgcn_wmma_f32_16x16x4_f32(
                false, a, false, bA, (short)0, accA, false, false);
            accB = __builtin_amdgcn_wmma_f32_16x16x4_f32(
                false, a, false, bB, (short)0, accB, false, false);
        }
        // C tile -> LDS: lane holds rows v + (lane<16 ? 0 : 8), col = e16
        {
            const int rbase = (lane < 16) ? 0 : 8;
#pragma unroll
            for (int v = 0; v < 8; ++v) {
                memv_lds[0][rbase + v][e16] = accA[v];
                memv_lds[1][rbase + v][e16] = accB[v];
            }
        }
        __syncthreads();

        // prefetch this chunk's input current (latency overlapped with GEMM epilogue)
        float Ireg[CHUNK];
#pragma unroll
        for (int r = 0; r < CHUNK; ++r)
            Ireg[r] = I[(size_t)(n0 + r) * BS + eg];

        // ===== sequential phase: 16 steps, each lane owns one neuron =====
#pragma unroll
        for (int r = 0; r < CHUNK; ++r) {
            const int n = n0 + r;
            float spike = (V - THRESHOLD_F >= 0.0f) ? 1.0f : 0.0f;
            float V_new;
            if (n == 0) {
                V_new = V_INIT_F;
                spike = 0.0f;
                d_lds[grp][0][e16] = 0.0f;
            } else if (n == 1) {
                V_new = V + DT_OVER_TAU_F * (-V + Ireg[r] * INV_GL_F);
                V_new = V_new - spike * (V_new - V_RESET_F);
                d_lds[grp][0][e16] = V_new - V;
            } else {
                // past chunks via WMMA + intra-chunk triangular tail
                float memV = memv_lds[grp][r][e16];
                const int jlo = (n0 > 0) ? (n0 - 1) : 0;
                for (int j = jlo; j <= n - 2; ++j)
                    memV += d_lds[grp][j][e16] * ktab[n - j];
                float V2 = GAMMA_C_F * (-GL_F * (V - VL_F) + Ireg[r]) * INV_CM_F
                           + V - memV;
                V_new = V2 - spike * (V2 - V_RESET_F);
                d_lds[grp][n - 1][e16] = V_new - V;
            }
            spikes_out[(size_t)n * BS + eg] = spike;
            V_out[(size_t)n * BS + eg]      = V_new;
            V = V_new;
        }
        __syncthreads();  // deltas visible before next chunk's GEMM reads
    }
}

extern "C" void kernel_launch(void* const* d_in, const int* in_sizes, int n_in,
                              void* d_out, int out_size, void* d_ws, size_t ws_size,
                              hipStream_t stream) {
    const float* I  = (const float*)d_in[0];   // [T,B,S] f32
    const float* V0 = (const float*)d_in[1];   // [B,S]   f32
    float* spikes = (float*)d_out;                         // [T,B,S]
    float* Vs     = spikes + (size_t)T_STEPS * BS;         // [T,B,S]
    dim3 grid(BS / 32);   // 1024 single-wave blocks, 32 neurons each
    flif_scan_kernel<<<grid, dim3(32), 0, stream>>>(I, V0, spikes, Vs);
}